// EverythingV9Engine_47098611368146
// MI455X (gfx1250) — compile-verified
//
#include <hip/hip_runtime.h>
#include <hip/hip_bf16.h>

typedef __attribute__((ext_vector_type(2))) float v2f;
typedef __attribute__((ext_vector_type(8))) float v8f;

constexpr int   NC   = 1024;      // n_cells
constexpr int   HD   = 1024;      // hidden_dim
constexpr int   DIN  = 1024;      // input_dim
constexpr int   NOBJ = 16;
constexpr int   MPO  = 4;
constexpr int   CPO  = NC / NOBJ; // 64
constexpr size_t NHE = (size_t)NC * HD;
constexpr float TWO_PI_F = 6.283185307179586f;
constexpr float PI_F     = 3.14159265358979f;

// ---------------- workspace layout (float offsets) ----------------
constexpr size_t OFF_CSRE = 0;
constexpr size_t OFF_CSIM = NHE;
constexpr size_t OFF_NEWRE = 2 * NHE;
constexpr size_t OFF_NEWIM = 3 * NHE;
constexpr size_t WB = 4 * NHE;
constexpr size_t OFF_EF     = WB + 0;        // [1024]
constexpr size_t OFF_SRCRE  = WB + 1024;     // [1024]
constexpr size_t OFF_SRCIM  = WB + 2048;     // [1024]
constexpr size_t OFF_TRRE   = WB + 3072;     // [4096]
constexpr size_t OFF_TRIM   = WB + 7168;     // [4096]
constexpr size_t OFF_OBJRE  = WB + 11264;    // [16384]
constexpr size_t OFF_OBJIM  = WB + 27648;    // [16384]
constexpr size_t OFF_MORRE  = WB + 44032;    // [65536]
constexpr size_t OFF_MORIM  = WB + 109568;   // [65536]
constexpr size_t OFF_CAVRE2 = WB + 175104;   // [1024]
constexpr size_t OFF_CAVIM2 = WB + 176128;   // [1024]
constexpr size_t OFF_CAVANG = WB + 177152;   // [1024]
constexpr size_t OFF_HUBM   = WB + 178176;   // [1024]
constexpr size_t OFF_COH    = WB + 179200;   // [1024]
constexpr size_t OFF_SCAL   = WB + 180224;   // [8]: 0=xmax 1=ampSum 2=ampSqSum 3=gateSum 4=catT

// ---------------- reduction helpers ----------------
__device__ __forceinline__ float wred_sum(float v) {
#pragma unroll
  for (int o = 16; o > 0; o >>= 1) v += __shfl_xor(v, o, 32);
  return v;
}
__device__ __forceinline__ float wred_max(float v) {
#pragma unroll
  for (int o = 16; o > 0; o >>= 1) v = fmaxf(v, __shfl_xor(v, o, 32));
  return v;
}
__device__ float bred_sum(float v, volatile float* sm) {
  __syncthreads();
  v = wred_sum(v);
  int lane = threadIdx.x & 31, w = threadIdx.x >> 5;
  if (lane == 0) sm[w] = v;
  __syncthreads();
  if (threadIdx.x == 0) {
    float r = 0.f;
    int nw = blockDim.x >> 5;
    for (int i = 0; i < nw; ++i) r += sm[i];
    sm[0] = r;
  }
  __syncthreads();
  float r = sm[0];
  return r;
}
__device__ float bred_max(float v, volatile float* sm) {
  __syncthreads();
  v = wred_max(v);
  int lane = threadIdx.x & 31, w = threadIdx.x >> 5;
  if (lane == 0) sm[w] = v;
  __syncthreads();
  if (threadIdx.x == 0) {
    float r = -3.4e38f;
    int nw = blockDim.x >> 5;
    for (int i = 0; i < nw; ++i) r = fmaxf(r, sm[i]);
    sm[0] = r;
  }
  __syncthreads();
  float r = sm[0];
  return r;
}

// ---------------- kernels ----------------
__global__ void k_init(float* scal) {
  if (threadIdx.x < 8) scal[threadIdx.x] = 0.f;
}

__global__ void k_xmax(const float* __restrict__ x, float* scal) {
  __shared__ float sm[32];
  int tid = threadIdx.x;
  float m = 0.f;
#pragma unroll
  for (int u = 0; u < 4; ++u) m = fmaxf(m, fabsf(x[tid + u * 256]));
  m = bred_max(m, sm);
  if (tid == 0) scal[0] = m;
}

// pump = sigmoid(x @ pump_w.T + b); ef = clip(exc*0.95 + pump*0.05, 0, 1)
__global__ void k_pump(const float* __restrict__ x, const float* __restrict__ pw,
                       const float* __restrict__ pb, const float* __restrict__ exc,
                       float* __restrict__ ef) {
  int row = (blockIdx.x * blockDim.x + threadIdx.x) >> 5;
  int lane = threadIdx.x & 31;
  const float* w = pw + (size_t)row * DIN;
  float s = 0.f;
  for (int j = lane; j < DIN; j += 32) s += x[j] * w[j];
  s = wred_sum(s);
  if (lane == 0) {
    float p = 1.f / (1.f + expf(-(s + pb[row])));
    ef[row] = fminf(fmaxf(exc[row] * 0.95f + p * 0.05f, 0.f), 1.f);
  }
}

// cs = (re + i im) * exp(i * 0.1 * pv)
__global__ void k_rotate(const float* __restrict__ re, const float* __restrict__ im,
                         const float* __restrict__ pv, float* __restrict__ csRe,
                         float* __restrict__ csIm) {
  size_t idx = (size_t)blockIdx.x * blockDim.x + threadIdx.x;
  float a = pv[idx] * 0.1f;
  float s, c;
  sincosf(a, &s, &c);
  float r = re[idx], i = im[idx];
  csRe[idx] = r * c - i * s;
  csIm[idx] = r * s + i * c;
}

// neighbor phase-difference coupling; new = 0.7 cs + 0.3*0.1/cnt * sum mask*cos(dang)*nb
__global__ void k_couple(const float* __restrict__ csRe, const float* __restrict__ csIm,
                         const int* __restrict__ nbrIdx, const float* __restrict__ nbrMask,
                         int K, float* __restrict__ newRe, float* __restrict__ newIm) {
  __shared__ int sJ[32];
  __shared__ float sM[32];
  __shared__ float sCnt;
  int i = blockIdx.x, tid = threadIdx.x;
  if (tid < K) { sJ[tid] = nbrIdx[i * K + tid]; sM[tid] = nbrMask[i * K + tid]; }
  __syncthreads();
  if (tid == 0) {
    float c = 0.f;
    for (int k = 0; k < K; ++k) c += sM[k];
    sCnt = fmaxf(c, 1.f);
  }
  __syncthreads();
  float inv = 0.03f / sCnt;  // 0.3 * 0.1 / cnt
#pragma unroll
  for (int u = 0; u < HD / 256; ++u) {
    int h = tid + u * 256;
    size_t p = (size_t)i * HD + h;
    float cr = csRe[p], ci = csIm[p];
    float ri = sqrtf(cr * cr + ci * ci);
    float ar = 0.f, ai = 0.f;
    for (int k = 0; k < K; ++k) {
      size_t jp = (size_t)sJ[k] * HD + h;
      float jr = csRe[jp], ji = csIm[jp];
      float rj = sqrtf(jr * jr + ji * ji);
      float den = ri * rj;
      float cosd = den > 1e-30f ? (cr * jr + ci * ji) / den : 0.f;
      float w = sM[k] * cosd;
      ar += w * jr;
      ai += w * ji;
    }
    newRe[p] = 0.7f * cr + inv * ar;
    newIm[p] = 0.7f * ci + inv * ai;
  }
}

// mean over the 64 cells of object o (oParam >= 0: single object -> out[0:H];
// oParam < 0: o = blockIdx.y, out offset o*H)
__global__ void k_objmean(const float* __restrict__ nRe, const float* __restrict__ nIm,
                          float* __restrict__ outRe, float* __restrict__ outIm, int oParam) {
  int o = (oParam >= 0) ? oParam : (int)blockIdx.y;
  int h = blockIdx.x * blockDim.x + threadIdx.x;
  float sr = 0.f, si = 0.f;
  size_t r0 = (size_t)o * CPO;
  for (int c = 0; c < CPO; ++c) {
    size_t p = (r0 + c) * HD + h;
    sr += nRe[p];
    si += nIm[p];
  }
  int off = (oParam >= 0) ? 0 : o * HD;
  outRe[off + h] = sr * (1.f / CPO);
  outIm[off + h] = si * (1.f / CPO);
}

// WMMA skinny GEMM: out[m*H + i] = sum_j W[m,i,j] * src[j] for m=0..3, re & im
// in one pass. A tile: 16 rows x K=4, B tile: K=4 x 16 (col0=re, col1=im, rest 0).
// V_WMMA_F32_16X16X4_F32, fp32-exact. One wave per 16-row tile, 256 K-steps.
// B fragment built branch-free: unconditional v2f loads + v_cndmask selects so
// EXEC stays all-1s through the loop (no saveexec sandwiches around the loads).
// Two accumulator chains break the WMMA D->C serial dependency; prefetch the
// A (weight) stream one 128B chunk ahead.
__global__ void k_wmma(const float* __restrict__ Wb, const float* __restrict__ srcRe,
                       const float* __restrict__ srcIm, float* __restrict__ outRe,
                       float* __restrict__ outIm, size_t wStride, int sStride, int oStride) {
  int o = blockIdx.y;
  const float* Wp = Wb + (size_t)o * wStride;
  const float* sR = srcRe + (size_t)o * sStride;
  const float* sI = srcIm + (size_t)o * sStride;
  float* oR = outRe + (size_t)o * oStride;
  float* oI = outIm + (size_t)o * oStride;

  int wave = (blockIdx.x * blockDim.x + threadIdx.x) >> 5;  // 0..255 (4096 rows / 16)
  int lane = threadIdx.x & 31;
  int sel = lane >> 4;    // K-half select (0: K=0,1 ; 1: K=2,3)
  int lrow = lane & 15;   // A row within tile / B column
  int r0 = wave * 16;

  const float* Arow = Wp + (size_t)(r0 + lrow) * HD + 2 * sel;
  const int kkOff = 2 * sel;
  bool isRe = (lrow == 0);
  bool isIm = (lrow == 1);

  v8f acc0 = {0.f, 0.f, 0.f, 0.f, 0.f, 0.f, 0.f, 0.f};
  v8f acc1 = {0.f, 0.f, 0.f, 0.f, 0.f, 0.f, 0.f, 0.f};

  for (int kb = 0; kb < HD; kb += 32) {
    __builtin_prefetch(Arow + kb + 32, 0, 3);  // global_prefetch_b8, next chunk of W
#pragma unroll
    for (int k0 = kb; k0 < kb + 32; k0 += 8) {
      {
        v2f a = *(const v2f*)(Arow + k0);
        int kk = k0 + kkOff;
        v2f br = *(const v2f*)(sR + kk);   // unconditional: in-bounds for all lanes
        v2f bi = *(const v2f*)(sI + kk);
        v2f b;
        b.x = isRe ? br.x : (isIm ? bi.x : 0.f);
        b.y = isRe ? br.y : (isIm ? bi.y : 0.f);
        acc0 = __builtin_amdgcn_wmma_f32_16x16x4_f32(false, a, false, b, (short)0, acc0,
                                                     false, false);
      }
      {
        v2f a = *(const v2f*)(Arow + k0 + 4);
        int kk = k0 + 4 + kkOff;
        v2f br = *(const v2f*)(sR + kk);
        v2f bi = *(const v2f*)(sI + kk);
        v2f b;
        b.x = isRe ? br.x : (isIm ? bi.x : 0.f);
        b.y = isRe ? br.y : (isIm ? bi.y : 0.f);
        acc1 = __builtin_amdgcn_wmma_f32_16x16x4_f32(false, a, false, b, (short)0, acc1,
                                                     false, false);
      }
    }
  }
  v8f acc = acc0 + acc1;

  // D layout: lane n<16 -> rows 0..7 of col n in v0..7; lane 16+n -> rows 8..15.
  if (lrow < 2) {
    float* dst = (lrow == 0) ? oR : oI;
    int rb = r0 + sel * 8;
#pragma unroll
    for (int v = 0; v < 8; ++v) dst[rb + v] = acc[v];
  }
}

// ns[tgt] = 0.95*ns[tgt] + 0.05*tr[m]  (tgt = (oi+m+1)%16, broadcast over 64 cells)
__global__ void k_scatter(float* __restrict__ nRe, float* __restrict__ nIm,
                          const float* __restrict__ trRe, const float* __restrict__ trIm,
                          int oi) {
  int idx = blockIdx.x * blockDim.x + threadIdx.x;  // < 4*64*1024
  int m = idx >> 16;
  int rem = idx & 0xFFFF;
  int c = rem >> 10;
  int h = rem & 1023;
  int tgt = (oi + m + 1) & 15;
  size_t p = ((size_t)(tgt * CPO + c)) * HD + h;
  nRe[p] = 0.95f * nRe[p] + 0.05f * trRe[m * HD + h];
  nIm[p] = 0.95f * nIm[p] + 0.05f * trIm[m * HD + h];
}

// cat_tension = mean_h |limit - colimit|^2
__global__ void k_cat(const float* __restrict__ objRe, const float* __restrict__ objIm,
                      const float* __restrict__ morRe, const float* __restrict__ morIm,
                      float* scal) {
  __shared__ float sm[32];
  int tid = threadIdx.x;
  float acc = 0.f;
#pragma unroll
  for (int u = 0; u < 4; ++u) {
    int h = tid + u * 256;
    float lr = 0.f, li = 0.f;
    for (int o = 0; o < NOBJ; ++o) { lr += objRe[o * HD + h]; li += objIm[o * HD + h]; }
    lr *= (1.f / NOBJ);
    li *= (1.f / NOBJ);
    float cr = 0.f, ci = 0.f;
    for (int q = 0; q < NOBJ * MPO; ++q) { cr += morRe[q * HD + h]; ci += morIm[q * HD + h]; }
    cr *= (1.f / (NOBJ * MPO));
    ci *= (1.f / (NOBJ * MPO));
    float dr = lr - cr, di = li - ci;
    acc += dr * dr + di * di;
  }
  acc = bred_sum(acc, sm);
  if (tid == 0) scal[4] = acc * (1.f / HD);
}

__global__ void k_cavang(const float* __restrict__ cr, const float* __restrict__ ci,
                         float* __restrict__ cavAng) {
  int h = blockIdx.x * blockDim.x + threadIdx.x;
  cavAng[h] = atan2f(ci[h], cr[h]);
}

// masked lasing phase-lock: where(ef>0.5) new = |new| * exp(i*(0.3*angle(cav)+0.7*angle(new)))
__global__ void k_phaselock(float* __restrict__ nRe, float* __restrict__ nIm,
                            const float* __restrict__ ef, const float* __restrict__ cavAng) {
  size_t idx = (size_t)blockIdx.x * blockDim.x + threadIdx.x;
  int i = (int)(idx >> 10);
  int h = (int)(idx & 1023);
  if (ef[i] > 0.5f) {
    float re = nRe[idx], im = nIm[idx];
    float r = sqrtf(re * re + im * im);
    float th = 0.3f * cavAng[h] + 0.7f * atan2f(im, re);
    float s, c;
    sincosf(th, &s, &c);
    nRe[idx] = r * c;
    nIm[idx] = r * s;
  }
}

// cav update: column reduce over lasing rows
__global__ void k_cavupd(const float* __restrict__ nRe, const float* __restrict__ nIm,
                         const float* __restrict__ ef, const float* __restrict__ cr,
                         const float* __restrict__ ci, float* __restrict__ cavRe2,
                         float* __restrict__ cavIm2) {
  int h = blockIdx.x * blockDim.x + threadIdx.x;
  float cnt = 0.f, sr = 0.f, si = 0.f;
  for (int i = 0; i < NC; ++i) {
    if (ef[i] > 0.5f) {
      cnt += 1.f;
      size_t p = (size_t)i * HD + h;
      sr += nRe[p];
      si += nIm[p];
    }
  }
  float inv = 1.f / fmaxf(cnt, 1.f);
  float c0 = cr[h], c1 = ci[h];
  cavRe2[h] = (cnt > 0.f) ? 0.8f * c0 + 0.2f * sr * inv : c0;
  cavIm2[h] = (cnt > 0.f) ? 0.8f * c1 + 0.2f * si * inv : c1;
}

// per-row: traveling wave, frustration regulation, input perturbation,
// amplitude normalization, per-cell coherence, amp-variance accumulation.
__global__ void k_row(const float* __restrict__ nRe, const float* __restrict__ nIm,
                      const float* __restrict__ noise, const float* __restrict__ x,
                      const int* __restrict__ stepPtr, float* scal,
                      float* __restrict__ amp, float* __restrict__ coh) {
  __shared__ float sm[32];
  int i = blockIdx.x, tid = threadIdx.x;
  float t = 0.1f * (float)(*stepPtr);
  float wavef = 1.f + 0.02f * sinf(t + TWO_PI_F * (float)i / (float)NC);
  float invx = 1.f / (scal[0] + 1e-8f);
  float strength = 0.1f / (1.f + 0.2f * (float)i);

  float r[4], a[4];
  float sc = 0.f, ss = 0.f, sa = 0.f;
#pragma unroll
  for (int u = 0; u < 4; ++u) {
    int h = tid + u * 256;
    size_t p = (size_t)i * HD + h;
    float re = nRe[p] * wavef, im = nIm[p] * wavef;
    float rr = sqrtf(re * re + im * im);
    float ang, cc, sn;
    if (rr > 1e-30f) { ang = atan2f(im, re); cc = re / rr; sn = im / rr; }
    else { ang = 0.f; cc = 1.f; sn = 0.f; }
    r[u] = rr;
    a[u] = ang;
    sc += cc; ss += sn; sa += ang;
  }
  sc = bred_sum(sc, sm);
  ss = bred_sum(ss, sm);
  sa = bred_sum(sa, sm);
  float mr = sc / HD, mi = ss / HD, meanAng = sa / HD;
  float frust = 1.f - sqrtf(mr * mr + mi * mi);
  float delta = frust - 0.5f;
  float scl = fminf(0.2f, fabsf(delta));
  float blend = fminf(0.15f, fabsf(delta) * 0.3f);

  float mx = 0.f;
#pragma unroll
  for (int u = 0; u < 4; ++u) {
    int h = tid + u * 256;
    size_t p = (size_t)i * HD + h;
    float a2;
    if (delta < -0.05f)      a2 = a[u] + noise[p] * scl;
    else if (delta > 0.05f)  a2 = a[u] * (1.f - blend) + meanAng * blend;
    else                     a2 = a[u];
    a[u] = a2 + (x[h] * invx) * (0.1f * PI_F) * strength;  // input perturbation
    mx = fmaxf(mx, r[u]);
  }
  mx = bred_max(mx, sm);
  float invr = 1.f / (mx + 1e-8f);

  float c2 = 0.f, s2 = 0.f, asum = 0.f, a2sum = 0.f;
#pragma unroll
  for (int u = 0; u < 4; ++u) {
    int h = tid + u * 256;
    size_t p = (size_t)i * HD + h;
    float an = r[u] * invr;
    amp[p] = an;
    asum += an;
    a2sum += an * an;
    float sn, cs;
    sincosf(a[u], &sn, &cs);
    c2 += cs;
    s2 += sn;
  }
  c2 = bred_sum(c2, sm);
  s2 = bred_sum(s2, sm);
  asum = bred_sum(asum, sm);
  a2sum = bred_sum(a2sum, sm);
  if (tid == 0) {
    float mcr = c2 / HD, mci = s2 / HD;
    coh[i] = sqrtf(mcr * mcr + mci * mci);
    atomicAdd(&scal[1], asum);
    atomicAdd(&scal[2], a2sum);
  }
}

// top-8 coherent cells -> hub_mean
__global__ void k_hub(const float* __restrict__ coh, const float* __restrict__ amp,
                      float* __restrict__ hubMean) {
  __shared__ float sCoh[NC];
  __shared__ float sVal[256];
  __shared__ int sIdx[256];
  __shared__ int sSel[8];
  int tid = threadIdx.x;
#pragma unroll
  for (int u = 0; u < 4; ++u) sCoh[tid + u * 256] = coh[tid + u * 256];
  __syncthreads();
  for (int r = 0; r < 8; ++r) {
    float bv = -1e30f;
    int bi = 0x7FFFFFFF;
    for (int u = 0; u < 4; ++u) {
      int h = tid + u * 256;
      float v = sCoh[h];
      if (v > bv) { bv = v; bi = h; }
    }
    sVal[tid] = bv;
    sIdx[tid] = bi;
    __syncthreads();
    for (int s = 128; s > 0; s >>= 1) {
      if (tid < s) {
        float ov = sVal[tid + s];
        int oi = sIdx[tid + s];
        if (ov > sVal[tid] || (ov == sVal[tid] && oi < sIdx[tid])) {
          sVal[tid] = ov;
          sIdx[tid] = oi;
        }
      }
      __syncthreads();
    }
    if (tid == 0) {
      sSel[r] = sIdx[0];
      sCoh[sIdx[0]] = -1e30f;
    }
    __syncthreads();
  }
#pragma unroll
  for (int u = 0; u < 4; ++u) {
    int h = tid + u * 256;
    float s = 0.f;
    for (int j = 0; j < 8; ++j) s += amp[(size_t)sSel[j] * HD + h];
    hubMean[h] = s * 0.125f;
  }
}

// gate[i] = sigmoid(hub_mean . gate_w[i] + b); accumulate sum for mean
__global__ void k_gate(const float* __restrict__ hm, const float* __restrict__ gw,
                       const float* __restrict__ gb, float* scal) {
  int row = (blockIdx.x * blockDim.x + threadIdx.x) >> 5;
  int lane = threadIdx.x & 31;
  const float* w = gw + (size_t)row * HD;
  float s = 0.f;
  for (int j = lane; j < HD; j += 32) s += hm[j] * w[j];
  s = wred_sum(s);
  if (lane == 0) {
    float g = 1.f / (1.f + expf(-(s + gb[row])));
    atomicAdd(&scal[3], g);
  }
}

// pred[d] = [gated.re, gated.im] . dec_w[d] + dec_b[d], gated = cav*gate_mean
__global__ void k_pred(const float* __restrict__ cavRe2, const float* __restrict__ cavIm2,
                       const float* __restrict__ scal, const float* __restrict__ dw,
                       const float* __restrict__ db, float* __restrict__ out) {
  int d = (blockIdx.x * blockDim.x + threadIdx.x) >> 5;
  int lane = threadIdx.x & 31;
  float gm = scal[3] * (1.f / NC);
  const float* w = dw + (size_t)d * (2 * HD);
  float s = 0.f;
  for (int k = lane; k < 2 * HD; k += 32) {
    float o = (k < HD) ? cavRe2[k] * gm : cavIm2[k - HD] * gm;
    s += w[k] * o;
  }
  s = wred_sum(s);
  if (lane == 0) out[d] = s + db[d];
}

__global__ void k_final(const float* scal, float* out) {
  float M = (float)NHE;
  float var = (scal[2] - scal[1] * scal[1] / M) / (M - 1.f);
  out[DIN] = scal[4] + var;
}

// ---------------- host ----------------
extern "C" void kernel_launch(void* const* d_in, const int* in_sizes, int n_in,
                              void* d_out, int out_size, void* d_ws, size_t ws_size,
                              hipStream_t stream) {
  (void)n_in; (void)out_size; (void)ws_size;
  const float* x        = (const float*)d_in[0];
  const int*   stepPtr  = (const int*)d_in[1];
  const int*   nbrIdx   = (const int*)d_in[2];
  const float* nbrMask  = (const float*)d_in[3];
  const float* cellRe   = (const float*)d_in[4];
  const float* cellIm   = (const float*)d_in[5];
  const float* phaseVel = (const float*)d_in[6];
  const float* exc      = (const float*)d_in[7];
  const float* cavRe    = (const float*)d_in[8];
  const float* cavIm    = (const float*)d_in[9];
  const float* morphW   = (const float*)d_in[10];
  const float* pumpW    = (const float*)d_in[11];
  const float* pumpB    = (const float*)d_in[12];
  const float* gateW    = (const float*)d_in[13];
  const float* gateB    = (const float*)d_in[14];
  const float* decW     = (const float*)d_in[15];
  const float* decB     = (const float*)d_in[16];
  const float* noise    = (const float*)d_in[17];
  float* out = (float*)d_out;
  float* ws = (float*)d_ws;
  int K = in_sizes[2] / NC;

  float* pCsRe = ws + OFF_CSRE;
  float* pCsIm = ws + OFF_CSIM;
  float* pNewRe = ws + OFF_NEWRE;
  float* pNewIm = ws + OFF_NEWIM;
  float* pEf = ws + OFF_EF;
  float* pSrcRe = ws + OFF_SRCRE;
  float* pSrcIm = ws + OFF_SRCIM;
  float* pTrRe = ws + OFF_TRRE;
  float* pTrIm = ws + OFF_TRIM;
  float* pObjRe = ws + OFF_OBJRE;
  float* pObjIm = ws + OFF_OBJIM;
  float* pMorRe = ws + OFF_MORRE;
  float* pMorIm = ws + OFF_MORIM;
  float* pCavRe2 = ws + OFF_CAVRE2;
  float* pCavIm2 = ws + OFF_CAVIM2;
  float* pCavAng = ws + OFF_CAVANG;
  float* pHubM = ws + OFF_HUBM;
  float* pCoh = ws + OFF_COH;
  float* pScal = ws + OFF_SCAL;
  float* pAmp = pCsRe;  // reuse cs buffer after morph stage

  k_init<<<1, 32, 0, stream>>>(pScal);
  k_xmax<<<1, 256, 0, stream>>>(x, pScal);
  k_pump<<<NC / 8, 256, 0, stream>>>(x, pumpW, pumpB, exc, pEf);
  k_rotate<<<(int)(NHE / 256), 256, 0, stream>>>(cellRe, cellIm, phaseVel, pCsRe, pCsIm);
  k_couple<<<NC, 256, 0, stream>>>(pCsRe, pCsIm, nbrIdx, nbrMask, K, pNewRe, pNewIm);

  // sequential categorical morphism scatter (WMMA GEMMs)
  for (int oi = 0; oi < NOBJ; ++oi) {
    k_objmean<<<dim3(HD / 256, 1), 256, 0, stream>>>(pNewRe, pNewIm, pSrcRe, pSrcIm, oi);
    k_wmma<<<dim3(32, 1), 256, 0, stream>>>(morphW + (size_t)oi * MPO * HD * HD, pSrcRe,
                                            pSrcIm, pTrRe, pTrIm, 0, 0, 0);
    k_scatter<<<(MPO * CPO * HD) / 256, 256, 0, stream>>>(pNewRe, pNewIm, pTrRe, pTrIm, oi);
  }
  k_objmean<<<dim3(HD / 256, NOBJ), 256, 0, stream>>>(pNewRe, pNewIm, pObjRe, pObjIm, -1);
  k_wmma<<<dim3(32, NOBJ), 256, 0, stream>>>(morphW, pObjRe, pObjIm, pMorRe, pMorIm,
                                             (size_t)MPO * HD * HD, HD, MPO * HD);
  k_cat<<<1, 256, 0, stream>>>(pObjRe, pObjIm, pMorRe, pMorIm, pScal);

  // lasing phase-lock + cavity update
  k_cavang<<<HD / 256, 256, 0, stream>>>(cavRe, cavIm, pCavAng);
  k_phaselock<<<(int)(NHE / 256), 256, 0, stream>>>(pNewRe, pNewIm, pEf, pCavAng);
  k_cavupd<<<HD / 256, 256, 0, stream>>>(pNewRe, pNewIm, pEf, cavRe, cavIm, pCavRe2, pCavIm2);

  // wave + frustration + perturbation + normalize + coherence
  k_row<<<NC, 256, 0, stream>>>(pNewRe, pNewIm, noise, x, stepPtr, pScal, pAmp, pCoh);

  // hub gate + decoder
  k_hub<<<1, 256, 0, stream>>>(pCoh, pAmp, pHubM);
  k_gate<<<NC / 8, 256, 0, stream>>>(pHubM, gateW, gateB, pScal);
  k_pred<<<DIN / 8, 256, 0, stream>>>(pCavRe2, pCavIm2, pScal, decW, decB, out);
  k_final<<<1, 1, 0, stream>>>(pScal, out);
}